// EdgeConv_33354716020955
// MI455X (gfx1250) — compile-verified
//
#include <hip/hip_runtime.h>
#include <hip/hip_bf16.h>

typedef float v2f __attribute__((ext_vector_type(2)));
typedef float v8f __attribute__((ext_vector_type(8)));

#define BB 8
#define CC 64
#define NN 4096
#define OUTC 64
#define KNN 20
#define STAT_BLOCKS 512
#define LRELU 0.2f
#define BN_EPS 1e-5f

// -------------------- Kernel 1: squared norms xx[b][n] --------------------
__global__ void k_norms(const float* __restrict__ x, float* __restrict__ xx) {
    int t = blockIdx.x * blockDim.x + threadIdx.x;   // 0 .. B*N-1
    if (t >= BB * NN) return;
    int b = t >> 12;          // /4096
    int n = t & (NN - 1);
    const float* xb = x + (size_t)b * CC * NN + n;
    float s = 0.f;
#pragma unroll 8
    for (int c = 0; c < CC; ++c) s += xb[(size_t)c * NN] * xb[(size_t)c * NN];
    xx[t] = s;
}

// -------------------- Kernel 2: fused distance GEMM (fp32 WMMA) + top-20 ----
// one wave (32 threads) per (batch, 16-row block)
__global__ void k_knn(const float* __restrict__ x, const float* __restrict__ xx,
                      int* __restrict__ idx) {
    const int lane = threadIdx.x;            // 0..31
    const int lm = lane & 15, lh = lane >> 4;
    const int b  = blockIdx.x >> 8;          // / (N/16)
    const int n0 = (blockIdx.x & 255) << 4;
    const float* xb = x + (size_t)b * CC * NN;
    const float* xxb = xx + (size_t)b * NN;

    // A tile: 16 rows (points n0..n0+15) x K=64, held for whole loop
    v2f a[16];
#pragma unroll
    for (int kc = 0; kc < 16; ++kc) {
        int kb = 4 * kc + 2 * lh;
        a[kc].x = xb[(size_t)kb * NN + n0 + lm];
        a[kc].y = xb[(size_t)(kb + 1) * NN + n0 + lm];
    }

    __shared__ float xxa[16];
    __shared__ float dtile[256];
    __shared__ float bestd[16][KNN];
    __shared__ int   besti[16][KNN];
    if (lane < 16) {
        xxa[lane] = xxb[n0 + lane];
        for (int j = 0; j < KNN; ++j) { bestd[lane][j] = 3.4e38f; besti[lane][j] = 0; }
    }
    __syncthreads();

    for (int m0 = 0; m0 < NN; m0 += 16) {
        v8f acc = {0.f, 0.f, 0.f, 0.f, 0.f, 0.f, 0.f, 0.f};
#pragma unroll
        for (int kc = 0; kc < 16; ++kc) {
            int kb = 4 * kc + 2 * lh;
            v2f bt;
            bt.x = xb[(size_t)kb * NN + m0 + lm];
            bt.y = xb[(size_t)(kb + 1) * NN + m0 + lm];
            acc = __builtin_amdgcn_wmma_f32_16x16x4_f32(
                false, a[kc], false, bt, (short)0, acc, false, false);
        }
        float xxcol = xxb[m0 + lm];
#pragma unroll
        for (int i = 0; i < 8; ++i) {
            int row = i + 8 * lh;                       // C/D layout: M = i + 8*(lane/16)
            float d = xxa[row] + xxcol - 2.f * acc[i];
            dtile[row * 16 + lm] = d;
        }
        __syncthreads();
        if (lane < 16) {
            int r = lane;
            float worst = bestd[r][KNN - 1];
            for (int j = 0; j < 16; ++j) {
                float d = dtile[r * 16 + j];
                if (d < worst) {
                    int p = KNN - 1;
                    while (p > 0 && bestd[r][p - 1] > d) {
                        bestd[r][p] = bestd[r][p - 1];
                        besti[r][p] = besti[r][p - 1];
                        --p;
                    }
                    bestd[r][p] = d;
                    besti[r][p] = m0 + j;
                    worst = bestd[r][KNN - 1];
                }
            }
        }
        __syncthreads();
    }
    if (lane < 16) {
        int* ip = idx + ((size_t)(b * NN + n0 + lane)) * KNN;
        for (int j = 0; j < KNN; ++j) ip[j] = besti[lane][j];
    }
}

// -------------------- Kernel 3: u = (W1-W2)^T x ; v = W2^T x (fp32 WMMA) ----
// one wave per (b, n-tile(16), o-tile(16))
__global__ void k_uv(const float* __restrict__ x, const float* __restrict__ W,
                     float* __restrict__ u, float* __restrict__ v) {
    const int lane = threadIdx.x;
    const int lm = lane & 15, lh = lane >> 4;
    int bid = blockIdx.x;                    // B * 256 * 4
    int b  = bid >> 10;
    int nt = (bid >> 2) & 255;
    int ot = bid & 3;
    int n0 = nt << 4, o0 = ot << 4;
    const float* xb = x + (size_t)b * CC * NN;

    v2f au[16], av[16], bt[16];
#pragma unroll
    for (int kc = 0; kc < 16; ++kc) {
        int kb = 4 * kc + 2 * lh;
        int o = o0 + lm;                             // A: M = lane%16
        float w1a = W[o * 128 + kb],      w1b = W[o * 128 + kb + 1];
        float w2a = W[o * 128 + 64 + kb], w2b = W[o * 128 + 64 + kb + 1];
        au[kc].x = w1a - w2a; au[kc].y = w1b - w2b;
        av[kc].x = w2a;       av[kc].y = w2b;
        bt[kc].x = xb[(size_t)kb * NN + n0 + lm];    // B: N = lane%16
        bt[kc].y = xb[(size_t)(kb + 1) * NN + n0 + lm];
    }
    v8f cu = {0.f, 0.f, 0.f, 0.f, 0.f, 0.f, 0.f, 0.f};
    v8f cv = {0.f, 0.f, 0.f, 0.f, 0.f, 0.f, 0.f, 0.f};
#pragma unroll
    for (int kc = 0; kc < 16; ++kc) {
        cu = __builtin_amdgcn_wmma_f32_16x16x4_f32(false, au[kc], false, bt[kc],
                                                   (short)0, cu, false, false);
        cv = __builtin_amdgcn_wmma_f32_16x16x4_f32(false, av[kc], false, bt[kc],
                                                   (short)0, cv, false, false);
    }
#pragma unroll
    for (int i = 0; i < 8; ++i) {
        int o = o0 + i + 8 * lh;
        int n = n0 + lm;
        size_t base = ((size_t)(b * NN + n)) * OUTC + o;
        u[base] = cu[i];
        v[base] = cv[i];
    }
}

// -------------------- Kernel 4: BN partial sums (deterministic) -------------
__global__ void k_stats(const float* __restrict__ u, const float* __restrict__ v,
                        const int* __restrict__ idx,
                        float* __restrict__ psum, float* __restrict__ psumsq) {
    const int R = BB * NN * KNN;
    int o = threadIdx.x & 63;
    int sub = threadIdx.x >> 6;                       // 0..3
    float ls = 0.f, lsq = 0.f;
    for (int r = blockIdx.x * 4 + sub; r < R; r += STAT_BLOCKS * 4) {
        int bn = r / KNN, k = r - bn * KNN;
        int b = bn >> 12;
        int m = idx[(size_t)bn * KNN + k];
        float y = u[(size_t)bn * OUTC + o] + v[((size_t)(b * NN + m)) * OUTC + o];
        ls += y; lsq += y * y;
    }
    __shared__ float ssum[256], ssq[256];
    ssum[threadIdx.x] = ls; ssq[threadIdx.x] = lsq;
    __syncthreads();
    if (threadIdx.x < 64) {
        float a = ssum[threadIdx.x] + ssum[threadIdx.x + 64] +
                  ssum[threadIdx.x + 128] + ssum[threadIdx.x + 192];
        float q = ssq[threadIdx.x] + ssq[threadIdx.x + 64] +
                  ssq[threadIdx.x + 128] + ssq[threadIdx.x + 192];
        psum[blockIdx.x * 64 + o]   = a;
        psumsq[blockIdx.x * 64 + o] = q;
    }
}

// -------------------- Kernel 5: finalize scale/shift ------------------------
__global__ void k_finalize(const float* __restrict__ psum, const float* __restrict__ psumsq,
                           const float* __restrict__ gamma, const float* __restrict__ beta,
                           float* __restrict__ scale, float* __restrict__ shift) {
    int o = threadIdx.x;
    if (o >= OUTC) return;
    float s = 0.f, q = 0.f;
    for (int i = 0; i < STAT_BLOCKS; ++i) { s += psum[i * 64 + o]; q += psumsq[i * 64 + o]; }
    const float invR = 1.0f / (float)(BB * NN * KNN);
    float mean = s * invR;
    float var  = q * invR - mean * mean;
    float inv  = rsqrtf(var + BN_EPS);
    float sc   = gamma[o] * inv;
    scale[o] = sc;
    shift[o] = beta[o] - mean * sc;
}

// -------------------- Kernel 6: normalize + LeakyReLU + max over k ----------
__global__ void k_out(const float* __restrict__ u, const float* __restrict__ v,
                      const int* __restrict__ idx,
                      const float* __restrict__ scale, const float* __restrict__ shift,
                      float* __restrict__ out) {
    int e = blockIdx.x * blockDim.x + threadIdx.x;    // 0 .. B*N*OUTC-1
    if (e >= BB * NN * OUTC) return;
    int bn = e >> 6;
    int o  = e & 63;
    int b  = bn >> 12;
    int n  = bn & (NN - 1);
    float ub = u[e];
    float sc = scale[o], sh = shift[o];
    const int* ip = idx + (size_t)bn * KNN;
    float mx = -3.4e38f;
#pragma unroll
    for (int k = 0; k < KNN; ++k) {
        int m = ip[k];
        float y = (ub + v[((size_t)(b * NN + m)) * OUTC + o]) * sc + sh;
        y = (y >= 0.f) ? y : LRELU * y;
        mx = fmaxf(mx, y);
    }
    out[(size_t)b * OUTC * NN + (size_t)o * NN + n] = mx;
}

extern "C" void kernel_launch(void* const* d_in, const int* in_sizes, int n_in,
                              void* d_out, int out_size, void* d_ws, size_t ws_size,
                              hipStream_t stream) {
    const float* x     = (const float*)d_in[0];   // (8,64,4096)
    const float* W     = (const float*)d_in[1];   // (64,128)
    const float* gamma = (const float*)d_in[2];   // (64,)
    const float* beta  = (const float*)d_in[3];   // (64,)
    float* out = (float*)d_out;                   // (8,64,4096)

    // workspace carve-up
    float* xx     = (float*)d_ws;                         // 32768
    int*   idx    = (int*)(xx + BB * NN);                 // 655360
    float* u      = (float*)(idx + BB * NN * KNN);        // 2097152
    float* v      = u + (size_t)BB * NN * OUTC;           // 2097152
    float* psum   = v + (size_t)BB * NN * OUTC;           // 512*64
    float* psumsq = psum + STAT_BLOCKS * 64;              // 512*64
    float* scale  = psumsq + STAT_BLOCKS * 64;            // 64
    float* shift  = scale + 64;                           // 64

    k_norms<<<(BB * NN + 255) / 256, 256, 0, stream>>>(x, xx);
    k_knn<<<BB * (NN / 16), 32, 0, stream>>>(x, xx, idx);
    k_uv<<<BB * (NN / 16) * (OUTC / 16), 32, 0, stream>>>(x, W, u, v);
    k_stats<<<STAT_BLOCKS, 256, 0, stream>>>(u, v, idx, psum, psumsq);
    k_finalize<<<1, 64, 0, stream>>>(psum, psumsq, gamma, beta, scale, shift);
    k_out<<<(BB * NN * OUTC + 255) / 256, 256, 0, stream>>>(u, v, idx, scale, shift, out);
}